// SAGE_13743895347313
// MI455X (gfx1250) — compile-verified
//
#include <hip/hip_runtime.h>
#include <hip/hip_bf16.h>

typedef float v2f __attribute__((ext_vector_type(2)));
typedef float v8f __attribute__((ext_vector_type(8)));

// ---------------- utility kernels ----------------

__global__ void zero_f32(float* __restrict__ p, long n) {
    long i = (long)blockIdx.x * blockDim.x + threadIdx.x;
    if (i < n) p[i] = 0.f;
}

__global__ void deg_kernel(const int* __restrict__ dst, float* __restrict__ deg, int E) {
    int e = blockIdx.x * blockDim.x + threadIdx.x;
    if (e < E) unsafeAtomicAdd(&deg[dst[e]], 1.0f);
}

__global__ void invdeg_kernel(float* __restrict__ d, int N) {
    int i = blockIdx.x * blockDim.x + threadIdx.x;
    if (i < N) d[i] = 1.0f / fmaxf(d[i], 1.0f);
}

// (edge, feature) parallel scatter-add: F = 17
__global__ void agg1_kernel(const float* __restrict__ x, const int* __restrict__ src,
                            const int* __restrict__ dst, float* __restrict__ agg, long total) {
    long i = (long)blockIdx.x * blockDim.x + threadIdx.x;
    if (i < total) {
        long e = i / 17;
        int  f = (int)(i - e * 17);
        unsafeAtomicAdd(&agg[(long)dst[e] * 17 + f], x[(long)src[e] * 17 + f]);
    }
}

// (edge, feature) parallel scatter-add: F = 128
__global__ void agg2_kernel(const float* __restrict__ h, const int* __restrict__ src,
                            const int* __restrict__ dst, float* __restrict__ agg, long total) {
    long i = (long)blockIdx.x * blockDim.x + threadIdx.x;
    if (i < total) {
        long e = i >> 7;
        int  f = (int)(i & 127);
        unsafeAtomicAdd(&agg[(long)dst[e] * 128 + f], h[(long)src[e] * 128 + f]);
    }
}

// ---------------- WMMA SAGE layer 1: [N,17] -> [N,128] ----------------
// block = 256 thr = 8 waves; block covers 16 node rows; wave w covers cols [16w,16w+16)
// A rows beyond N are loaded from row 0 (clamped pointer) — harmless, their D rows
// are never stored, so no load guards and no EXEC divergence before the WMMAs.
__global__ __launch_bounds__(256) void sage_l1(
    const float* __restrict__ x, const float* __restrict__ agg1,
    const float* __restrict__ invdeg, const float* __restrict__ Ws1,
    const float* __restrict__ Wn1, const float* __restrict__ b1,
    float* __restrict__ h1, int N)
{
    int lane = threadIdx.x & 31;
    int wave = threadIdx.x >> 5;
    int r0   = blockIdx.x * 16;
    int col0 = wave * 16;
    int hs   = lane >> 4;       // half-select (K offset 0 or 2)
    int ls   = lane & 15;
    int arow = r0 + ls;         // A-matrix row this lane supplies
    int crow = arow < N ? arow : 0;  // clamped (safety only)
    int bcol = col0 + ls;       // B-matrix / output column this lane supplies
    const float* xr = x    + (long)crow * 17;
    const float* ar = agg1 + (long)crow * 17;
    float inv = invdeg[crow];

    v8f acc = {};
    // ---- self term: K = 0..15, unguarded ----
#pragma unroll
    for (int k = 0; k < 16; k += 4) {
        int ka = k + 2 * hs;
        v2f A = { xr[ka], xr[ka + 1] };
        v2f B = { Ws1[(long)ka * 128 + bcol], Ws1[(long)(ka + 1) * 128 + bcol] };
        acc = __builtin_amdgcn_wmma_f32_16x16x4_f32(false, A, false, B,
                                                    (short)0, acc, false, false);
    }
    // ---- self tail: K = 16 (only hs==0 lanes carry data; select, no branch) ----
    {
        float av = xr[16];
        float bv = Ws1[(long)16 * 128 + bcol];
        v2f A = { hs ? 0.f : av, 0.f };
        v2f B = { hs ? 0.f : bv, 0.f };
        acc = __builtin_amdgcn_wmma_f32_16x16x4_f32(false, A, false, B,
                                                    (short)0, acc, false, false);
    }
    // ---- mean-neighbor term: K = 0..15 ----
#pragma unroll
    for (int k = 0; k < 16; k += 4) {
        int ka = k + 2 * hs;
        v2f A = { ar[ka] * inv, ar[ka + 1] * inv };
        v2f B = { Wn1[(long)ka * 128 + bcol], Wn1[(long)(ka + 1) * 128 + bcol] };
        acc = __builtin_amdgcn_wmma_f32_16x16x4_f32(false, A, false, B,
                                                    (short)0, acc, false, false);
    }
    // ---- neighbor tail: K = 16 ----
    {
        float av = ar[16] * inv;
        float bv = Wn1[(long)16 * 128 + bcol];
        v2f A = { hs ? 0.f : av, 0.f };
        v2f B = { hs ? 0.f : bv, 0.f };
        acc = __builtin_amdgcn_wmma_f32_16x16x4_f32(false, A, false, B,
                                                    (short)0, acc, false, false);
    }

    float bias = b1[bcol];
#pragma unroll
    for (int g = 0; g < 8; g++) {
        int row = r0 + g + 8 * hs;
        if (row < N) h1[(long)row * 128 + bcol] = acc[g] + bias;
    }
}

// ---------------- WMMA SAGE layer 2: [N,128] -> [N,64] ----------------
// block = 128 thr = 4 waves; 16 node rows per block; wave w covers cols [16w,16w+16)
__global__ __launch_bounds__(128) void sage_l2(
    const float* __restrict__ h1, const float* __restrict__ agg2,
    const float* __restrict__ invdeg, const float* __restrict__ Ws2,
    const float* __restrict__ Wn2, const float* __restrict__ b2,
    float* __restrict__ h2, int N)
{
    int lane = threadIdx.x & 31;
    int wave = threadIdx.x >> 5;
    int r0   = blockIdx.x * 16;
    int col0 = wave * 16;
    int hs   = lane >> 4;
    int ls   = lane & 15;
    int arow = r0 + ls;
    int crow = arow < N ? arow : 0;
    int bcol = col0 + ls;
    const float* hr = h1   + (long)crow * 128;
    const float* ar = agg2 + (long)crow * 128;
    float inv = invdeg[crow];

    v8f acc = {};
#pragma unroll 4
    for (int k = 0; k < 128; k += 4) {
        int ka = k + 2 * hs;
        v2f A = { hr[ka], hr[ka + 1] };
        v2f B = { Ws2[(long)ka * 64 + bcol], Ws2[(long)(ka + 1) * 64 + bcol] };
        acc = __builtin_amdgcn_wmma_f32_16x16x4_f32(false, A, false, B,
                                                    (short)0, acc, false, false);
        v2f A2 = { ar[ka] * inv, ar[ka + 1] * inv };
        v2f B2 = { Wn2[(long)ka * 64 + bcol], Wn2[(long)(ka + 1) * 64 + bcol] };
        acc = __builtin_amdgcn_wmma_f32_16x16x4_f32(false, A2, false, B2,
                                                    (short)0, acc, false, false);
    }
    float bias = b2[bcol];
#pragma unroll
    for (int g = 0; g < 8; g++) {
        int row = r0 + g + 8 * hs;
        if (row < N) h2[(long)row * 64 + bcol] = acc[g] + bias;
    }
}

// ---------------- MLP head: 64 -> 32 -> 16 -> 1 (sigmoid) ----------------
__global__ __launch_bounds__(256) void mlp_kernel(
    const float* __restrict__ h2,
    const float* __restrict__ W3, const float* __restrict__ b3,
    const float* __restrict__ W4, const float* __restrict__ b4,
    const float* __restrict__ W5, const float* __restrict__ b5,
    float* __restrict__ out, int N)
{
    __shared__ float sW3[64 * 32];
    __shared__ float sW4[32 * 16];
    __shared__ float sb3[32];
    __shared__ float sb4[16];
    __shared__ float sW5[16];
    int tid = threadIdx.x;
    for (int i = tid; i < 64 * 32; i += 256) sW3[i] = W3[i];
    for (int i = tid; i < 32 * 16; i += 256) sW4[i] = W4[i];
    if (tid < 32) sb3[tid] = b3[tid];
    if (tid < 16) { sb4[tid] = b4[tid]; sW5[tid] = W5[tid]; }
    __syncthreads();

    int node = blockIdx.x * 256 + tid;
    if (node >= N) return;
    const float* h = h2 + (long)node * 64;

    float hr[64];
#pragma unroll
    for (int k = 0; k < 64; k++) hr[k] = h[k];

    float t3[32];
#pragma unroll
    for (int j = 0; j < 32; j++) {
        float s = sb3[j];
#pragma unroll
        for (int k = 0; k < 64; k++) s += hr[k] * sW3[k * 32 + j];
        t3[j] = fmaxf(s, 0.f);
    }
    float t4[16];
#pragma unroll
    for (int j = 0; j < 16; j++) {
        float s = sb4[j];
#pragma unroll
        for (int k = 0; k < 32; k++) s += t3[k] * sW4[k * 16 + j];
        t4[j] = fmaxf(s, 0.f);
    }
    float z = b5[0];
#pragma unroll
    for (int k = 0; k < 16; k++) z += t4[k] * sW5[k];
    out[node] = 1.f / (1.f + expf(-z));
}

// ---------------- host launcher ----------------

extern "C" void kernel_launch(void* const* d_in, const int* in_sizes, int n_in,
                              void* d_out, int out_size, void* d_ws, size_t ws_size,
                              hipStream_t stream)
{
    const float* x   = (const float*)d_in[0];
    const int*   src = (const int*)d_in[1];
    const int*   dst = (const int*)d_in[2];
    const float* Ws1 = (const float*)d_in[3];
    const float* Wn1 = (const float*)d_in[4];
    const float* b1  = (const float*)d_in[5];
    const float* Ws2 = (const float*)d_in[6];
    const float* Wn2 = (const float*)d_in[7];
    const float* b2  = (const float*)d_in[8];
    const float* W3  = (const float*)d_in[9];
    const float* b3  = (const float*)d_in[10];
    const float* W4  = (const float*)d_in[11];
    const float* b4  = (const float*)d_in[12];
    const float* W5  = (const float*)d_in[13];
    const float* b5  = (const float*)d_in[14];
    float* out = (float*)d_out;

    int N = in_sizes[0] / 17;
    int E = in_sizes[1];

    // workspace layout: invdeg[N] | agg1[N*17] | agg2[N*128] | h1[N*128] | h2[N*64]
    float* ws     = (float*)d_ws;
    float* invdeg = ws;
    float* agg1   = invdeg + N;
    float* agg2   = agg1 + (long)N * 17;
    float* h1     = agg2 + (long)N * 128;
    float* h2     = h1 + (long)N * 128;

    // 1) zero accumulators (invdeg + agg1 + agg2 are contiguous: N*146 floats)
    long nz = (long)N * 146;
    zero_f32<<<(unsigned)((nz + 255) / 256), 256, 0, stream>>>(ws, nz);

    // 2) degrees
    deg_kernel<<<(E + 255) / 256, 256, 0, stream>>>(dst, invdeg, E);

    // 3) layer-1 neighbor aggregation (edge x 17 atomics)
    long na1 = (long)E * 17;
    agg1_kernel<<<(unsigned)((na1 + 255) / 256), 256, 0, stream>>>(x, src, dst, agg1, na1);

    // 4) invdeg = 1/max(deg,1)
    invdeg_kernel<<<(N + 255) / 256, 256, 0, stream>>>(invdeg, N);

    // 5) SAGE layer 1 (WMMA)
    sage_l1<<<(N + 15) / 16, 256, 0, stream>>>(x, agg1, invdeg, Ws1, Wn1, b1, h1, N);

    // 6) layer-2 neighbor aggregation (edge x 128 atomics, served by 192MB L2)
    long na2 = (long)E * 128;
    agg2_kernel<<<(unsigned)((na2 + 255) / 256), 256, 0, stream>>>(h1, src, dst, agg2, na2);

    // 7) SAGE layer 2 (WMMA)
    sage_l2<<<(N + 15) / 16, 128, 0, stream>>>(h1, agg2, invdeg, Ws2, Wn2, b2, h2, N);

    // 8) MLP head + sigmoid
    mlp_kernel<<<(N + 255) / 256, 256, 0, stream>>>(h2, W3, b3, W4, b4, W5, b5, out, N);
}